// JointEndPoseGraphNet_50672024158729
// MI455X (gfx1250) — compile-verified
//
#include <hip/hip_runtime.h>
#include <hip/hip_bf16.h>
#include <math.h>

#define HID   128
#define ED    9
#define HEADS 4
#define OUTW  (HEADS * HID)   // 512
#define NEG_SLOPE 0.2f
#define EPS_LN 1e-5f
#define EPS_SM 1e-16f

typedef __attribute__((ext_vector_type(2))) float v2f;
typedef __attribute__((ext_vector_type(8))) float v8f;

// ---- monotone float <-> uint key (for atomicMax on signed floats) ----
__device__ __forceinline__ unsigned fkey(float f) {
    unsigned b = __float_as_uint(f);
    return (b & 0x80000000u) ? ~b : (b | 0x80000000u);
}
__device__ __forceinline__ float fdec(unsigned k) {
    return __uint_as_float((k & 0x80000000u) ? (k ^ 0x80000000u) : ~k);
}
#define NEG_INF_KEY 0x007FFFFFu   // fkey(-inf)

// ---------------------------------------------------------------------
// 1) x = concat(joint_features, endpose @ W_ep + b_ep)   [Ntot, 128]
// ---------------------------------------------------------------------
__global__ void k_build_x(const float* __restrict__ jf,
                          const float* __restrict__ eps,
                          const float* __restrict__ Wep,
                          const float* __restrict__ bep,
                          float* __restrict__ x, int nj, int Ntot) {
    int i = blockIdx.x * blockDim.x + threadIdx.x;
    if (i >= Ntot * HID) return;
    int n = i / HID, c = i % HID;
    if (n < nj) {
        x[i] = jf[i];
    } else {
        int f = n - nj;
        float acc = bep[c];
#pragma unroll
        for (int d = 0; d < ED; ++d) acc += eps[f * ED + d] * Wep[d * HID + c];
        x[i] = acc;
    }
}

// ---------------------------------------------------------------------
// 2) h = x @ W_gat   (M=Ntot, K=128, N=512)  via V_WMMA_F32_16X16X4_F32
//    block = 256 threads (8 waves); wave computes a 16x64 strip.
//    B strip staged in LDS *pair-interleaved*: Bs[kp][c][{k even,k odd}]
//    with kp-stride 160 floats so the two wave halves (kp vs kp+1) hit
//    disjoint 32-bank windows -> conflict-free ds_load_b64 fragments.
// ---------------------------------------------------------------------
#define KPSTRIDE 160   // floats per kp row in LDS (160 % 64 == 32)

__global__ void k_gemm_wmma(const float* __restrict__ X,
                            const float* __restrict__ W,
                            float* __restrict__ H, int Ntot) {
    __shared__ float Bs[64 * KPSTRIDE];   // 40 KB

    const int col0 = blockIdx.y * 64;
    // cooperative fill: Bs[(k>>1)*KPSTRIDE + c*2 + (k&1)] = W[k*512 + col0 + c]
    for (int idx = threadIdx.x; idx < 128 * 64; idx += 256) {
        int k = idx >> 6, c = idx & 63;
        Bs[(k >> 1) * KPSTRIDE + c * 2 + (k & 1)] = W[(size_t)k * OUTW + col0 + c];
    }
    __syncthreads();

    const int wave = threadIdx.x >> 5;
    const int lane = threadIdx.x & 31;
    const int row0 = (blockIdx.x * 8 + wave) * 16;

    // A-fragment addressing per ISA: lanes 0-15 hold M=lane,    K = k+0/k+1
    //                                lanes 16-31 hold M=lane-16, K = k+2/k+3
    // Clamp OOB rows instead of predicating loads: their D rows are masked
    // at the store, so reading a valid neighbor row is harmless and keeps
    // the inner loop branch-free (straight global_load_b64).
    const int mrow  = row0 + (lane & 15);
    const int mrowC = mrow < Ntot ? mrow : (Ntot - 1);
    const int ksel  = (lane >> 4) << 1;     // 0 or 2
    const float* xr = X + (size_t)mrowC * HID;

    const int cb = lane & 15;
    const float* bsBase = &Bs[cb * 2];

    v8f acc0 = {}, acc1 = {}, acc2 = {}, acc3 = {};

#pragma unroll
    for (int k = 0; k < HID; k += 4) {
        // one 8B global load per lane per step (k+ksel is even -> aligned)
        v2f a = *(const v2f*)(xr + k + ksel);

        const float* bk = bsBase + ((k + ksel) >> 1) * KPSTRIDE;
        v2f b0 = *(const v2f*)(bk +  0 * 32);
        v2f b1 = *(const v2f*)(bk +  1 * 32);
        v2f b2 = *(const v2f*)(bk +  2 * 32);
        v2f b3 = *(const v2f*)(bk +  3 * 32);

        acc0 = __builtin_amdgcn_wmma_f32_16x16x4_f32(false, a, false, b0, (short)0, acc0, false, false);
        acc1 = __builtin_amdgcn_wmma_f32_16x16x4_f32(false, a, false, b1, (short)0, acc1, false, false);
        acc2 = __builtin_amdgcn_wmma_f32_16x16x4_f32(false, a, false, b2, (short)0, acc2, false, false);
        acc3 = __builtin_amdgcn_wmma_f32_16x16x4_f32(false, a, false, b3, (short)0, acc3, false, false);
    }

    // C/D layout: VGPR j, lane L -> row = j + (L>=16 ? 8 : 0), col = L%16
    const int rowAdd  = (lane >> 4) << 3;
    const int colBase = col0 + (lane & 15);
#pragma unroll
    for (int j = 0; j < 8; ++j) {
        int r = row0 + rowAdd + j;
        if (r < Ntot) {
            float* hr = H + (size_t)r * OUTW + colBase;
            hr[ 0] = acc0[j];
            hr[16] = acc1[j];
            hr[32] = acc2[j];
            hr[48] = acc3[j];
        }
    }
}

// ---------------------------------------------------------------------
// 3) a_src[n,h] = <h[n,h,:], att_src[h,:]>  (and a_dst)  — wave per node
// ---------------------------------------------------------------------
__global__ void k_attn(const float* __restrict__ H,
                       const float* __restrict__ att_s,
                       const float* __restrict__ att_d,
                       float* __restrict__ a_s, float* __restrict__ a_d, int Ntot) {
    int wave = (blockIdx.x * blockDim.x + threadIdx.x) >> 5;
    int lane = threadIdx.x & 31;
    if (wave >= Ntot) return;
    const float* hrow = H + (size_t)wave * OUTW;
#pragma unroll
    for (int head = 0; head < HEADS; ++head) {
        float s = 0.f, d = 0.f;
#pragma unroll
        for (int i = 0; i < HID / 32; ++i) {
            int c = lane + i * 32;
            float hv = hrow[head * HID + c];
            s += hv * att_s[head * HID + c];
            d += hv * att_d[head * HID + c];
        }
#pragma unroll
        for (int off = 16; off; off >>= 1) {
            s += __shfl_xor(s, off, 32);
            d += __shfl_xor(d, off, 32);
        }
        if (lane == 0) {
            a_s[wave * HEADS + head] = s;
            a_d[wave * HEADS + head] = d;
        }
    }
}

// ---------------------------------------------------------------------
// 4) init accumulators
// ---------------------------------------------------------------------
__global__ void k_init(unsigned* __restrict__ amax, float* __restrict__ denom,
                       float* __restrict__ outacc, int Ntot) {
    int i = blockIdx.x * blockDim.x + threadIdx.x;
    if (i < Ntot * HEADS) { amax[i] = NEG_INF_KEY; denom[i] = 0.f; }
    if (i < Ntot * HID)   outacc[i] = 0.f;
}

// ---------------------------------------------------------------------
// 5) edge pass A: leaky-relu logits + segment max (key-encoded atomicMax)
//    thread per (edge, head); edges >= E are self-loops.
// ---------------------------------------------------------------------
__global__ void k_edge_max(const int* __restrict__ edg,
                           const float* __restrict__ a_s,
                           const float* __restrict__ a_d,
                           float* __restrict__ alpha, unsigned* __restrict__ amax,
                           int E, int Ntot) {
    int i = blockIdx.x * blockDim.x + threadIdx.x;
    if (i >= (E + Ntot) * HEADS) return;
    int e = i >> 2, head = i & 3;
    int s, d;
    if (e < E) { s = edg[e]; d = edg[E + e]; } else { s = d = e - E; }
    float v = a_s[s * HEADS + head] + a_d[d * HEADS + head];
    v = (v > 0.f) ? v : NEG_SLOPE * v;
    alpha[i] = v;
    atomicMax(&amax[d * HEADS + head], fkey(v));
}

// ---------------------------------------------------------------------
// 6) edge pass B: exp(alpha - max) + segment-sum denominators
// ---------------------------------------------------------------------
__global__ void k_edge_exp(const int* __restrict__ edg,
                           float* __restrict__ alpha,
                           const unsigned* __restrict__ amax,
                           float* __restrict__ denom, int E, int Ntot) {
    int i = blockIdx.x * blockDim.x + threadIdx.x;
    if (i >= (E + Ntot) * HEADS) return;
    int e = i >> 2, head = i & 3;
    int d = (e < E) ? edg[E + e] : (e - E);
    float ex = __expf(alpha[i] - fdec(amax[d * HEADS + head]));
    alpha[i] = ex;
    atomicAdd(&denom[d * HEADS + head], ex);
}

// ---------------------------------------------------------------------
// 7) edge pass C: out[dst] += (1/H) * sum_h alpha_norm[h] * h[src,h,:]
//    wave per edge; lanes sweep 128 channels (coalesced), 4 atomics/lane.
// ---------------------------------------------------------------------
__global__ void k_edge_agg(const int* __restrict__ edg,
                           const float* __restrict__ H,
                           const float* __restrict__ alpha,
                           const float* __restrict__ denom,
                           float* __restrict__ outacc, int E, int Ntot) {
    int wave = (blockIdx.x * blockDim.x + threadIdx.x) >> 5;
    int lane = threadIdx.x & 31;
    if (wave >= E + Ntot) return;
    int s, d;
    if (wave < E) { s = edg[wave]; d = edg[E + wave]; } else { s = d = wave - E; }

    float w[HEADS];
#pragma unroll
    for (int h = 0; h < HEADS; ++h)
        w[h] = alpha[(size_t)wave * HEADS + h] /
               (denom[d * HEADS + h] + EPS_SM) * (1.0f / HEADS);

    const float* hs = H + (size_t)s * OUTW;
    float* od = outacc + (size_t)d * HID;
#pragma unroll
    for (int i = 0; i < HID / 32; ++i) {
        int c = lane + i * 32;
        float acc = w[0] * hs[c] + w[1] * hs[HID + c] +
                    w[2] * hs[2 * HID + c] + w[3] * hs[3 * HID + c];
        atomicAdd(&od[c], acc);
    }
}

// ---------------------------------------------------------------------
// 8) bias + LayerNorm, write first NJ rows.  wave per row.
// ---------------------------------------------------------------------
__global__ void k_ln(const float* __restrict__ outacc,
                     const float* __restrict__ bg,
                     const float* __restrict__ gamma,
                     const float* __restrict__ beta,
                     float* __restrict__ out, int nj) {
    int wave = (blockIdx.x * blockDim.x + threadIdx.x) >> 5;
    int lane = threadIdx.x & 31;
    if (wave >= nj) return;
    const float* row = outacc + (size_t)wave * HID;

    float v[4], sum = 0.f, sq = 0.f;
#pragma unroll
    for (int i = 0; i < 4; ++i) {
        int c = lane + i * 32;
        v[i] = row[c] + bg[c];
        sum += v[i];
        sq  += v[i] * v[i];
    }
#pragma unroll
    for (int off = 16; off; off >>= 1) {
        sum += __shfl_xor(sum, off, 32);
        sq  += __shfl_xor(sq,  off, 32);
    }
    float mu  = sum * (1.0f / HID);
    float var = sq * (1.0f / HID) - mu * mu;
    float inv = rsqrtf(var + EPS_LN);
#pragma unroll
    for (int i = 0; i < 4; ++i) {
        int c = lane + i * 32;
        out[(size_t)wave * HID + c] = (v[i] - mu) * inv * gamma[c] + beta[c];
    }
}

// ---------------------------------------------------------------------
extern "C" void kernel_launch(void* const* d_in, const int* in_sizes, int n_in,
                              void* d_out, int out_size, void* d_ws, size_t ws_size,
                              hipStream_t stream) {
    const float* jf   = (const float*)d_in[0];
    const float* eps  = (const float*)d_in[1];
    const int*   edg  = (const int*)d_in[2];
    const float* Wep  = (const float*)d_in[3];
    const float* bep  = (const float*)d_in[4];
    const float* Wgat = (const float*)d_in[5];
    const float* attS = (const float*)d_in[6];
    const float* attD = (const float*)d_in[7];
    const float* bg   = (const float*)d_in[8];
    const float* gam  = (const float*)d_in[9];
    const float* bet  = (const float*)d_in[10];
    float* out = (float*)d_out;

    const int nj   = in_sizes[0] / HID;
    const int nf   = in_sizes[1] / ED;
    const int E    = in_sizes[2] / 2;
    const int Ntot = nj + nf;

    // workspace carve-out (fp32 elements)
    float* ws = (float*)d_ws;
    size_t o = 0;
    float*    x      = ws + o; o += (size_t)Ntot * HID;          // 25.6 MB
    float*    h      = ws + o; o += (size_t)Ntot * OUTW;         // 102.4 MB
    float*    a_s    = ws + o; o += (size_t)Ntot * HEADS;
    float*    a_d    = ws + o; o += (size_t)Ntot * HEADS;
    unsigned* amax   = (unsigned*)(ws + o); o += (size_t)Ntot * HEADS;
    float*    denom  = ws + o; o += (size_t)Ntot * HEADS;
    float*    alpha  = ws + o; o += (size_t)(E + Ntot) * HEADS;  // 4.8 MB
    float*    outacc = ws + o; o += (size_t)Ntot * HID;          // 25.6 MB

    // 1) build x
    {
        int n = Ntot * HID;
        k_build_x<<<(n + 255) / 256, 256, 0, stream>>>(jf, eps, Wep, bep, x, nj, Ntot);
    }
    // 2) WMMA GEMM: grid.x covers 128 rows/block, grid.y covers 64-col strips
    {
        dim3 grid((Ntot + 127) / 128, OUTW / 64);
        k_gemm_wmma<<<grid, 256, 0, stream>>>(x, Wgat, h, Ntot);
    }
    // 3) attention logits per node
    k_attn<<<(Ntot + 7) / 8, 256, 0, stream>>>(h, attS, attD, a_s, a_d, Ntot);
    // 4) init accumulators
    {
        int n = Ntot * HID;
        k_init<<<(n + 255) / 256, 256, 0, stream>>>(amax, denom, outacc, Ntot);
    }
    // 5/6) segment softmax over destinations
    {
        int n = (E + Ntot) * HEADS;
        k_edge_max<<<(n + 255) / 256, 256, 0, stream>>>(edg, a_s, a_d, alpha, amax, E, Ntot);
        k_edge_exp<<<(n + 255) / 256, 256, 0, stream>>>(edg, alpha, amax, denom, E, Ntot);
    }
    // 7) weighted aggregation (head-mean folded in)
    {
        int waves = E + Ntot;
        k_edge_agg<<<(waves + 7) / 8, 256, 0, stream>>>(edg, h, alpha, denom, outacc, E, Ntot);
    }
    // 8) bias + LayerNorm on joint rows
    k_ln<<<(nj + 7) / 8, 256, 0, stream>>>(outacc, bg, gam, bet, out, nj);
}